// MergeRec_56255481643081
// MI455X (gfx1250) — compile-verified
//
#include <hip/hip_runtime.h>
#include <hip/hip_bf16.h>

typedef __attribute__((ext_vector_type(2))) float v2f;
typedef __attribute__((ext_vector_type(8))) float v8f;

#define INV_S 0.08838834764831845f   /* 1/sqrt(128) */

// ws layout (floats): [0,16384) Wt_own (256x64), [16384,32768) Wt_ref (256x64),
//                     [32768,32832) v[64], [32832] c, [32833] s_hidden
#define WS_WTOWN 0
#define WS_WTREF 16384
#define WS_V     32768
#define WS_C     32832
#define WS_SHID  32833

__device__ __forceinline__ v8f wmma_f32(v2f a, v2f b, v8f c) {
  // V_WMMA_F32_16X16X4_F32 : D(16x16 f32) = A(16x4 f32) x B(4x16 f32) + C
  return __builtin_amdgcn_wmma_f32_16x16x4_f32(false, a, false, b, (short)0, c,
                                               false, false);
}

__global__ void prep_kernel(const float* __restrict__ own_w,   // 64x256
                            const float* __restrict__ ref_w,   // 64x256
                            const float* __restrict__ ew,      // 64x256 (enemy)
                            const float* __restrict__ eb,      // 256
                            const float* __restrict__ hidden,  // 64
                            float* __restrict__ ws) {
  int t = blockIdx.x * blockDim.x + threadIdx.x;  // 64*256 = 16384 threads
  if (t < 16384) {
    int n = t >> 6, k = t & 63;                   // Wt[n][k] = W[k][n]
    ws[WS_WTOWN + t] = own_w[k * 256 + n];
    ws[WS_WTREF + t] = ref_w[k * 256 + n];
  }
  if (blockIdx.x == 0) {
    __shared__ float k0[128];
    __shared__ float vsh[64];
    for (int a = threadIdx.x; a < 128; a += blockDim.x) {
      float s = eb[128 + a];
      for (int d = 0; d < 64; ++d) s += hidden[d] * ew[d * 256 + 128 + a];
      k0[a] = s;
    }
    __syncthreads();
    for (int d = threadIdx.x; d < 64; d += blockDim.x) {
      float s = 0.f;
      for (int a = 0; a < 128; ++a) s += ew[d * 256 + a] * k0[a];
      vsh[d] = s;
      ws[WS_V + d] = s;
    }
    __syncthreads();
    if (threadIdx.x == 0) {
      float c = 0.f;
      for (int a = 0; a < 128; ++a) c += eb[a] * k0[a];
      float hv = 0.f;
      for (int d = 0; d < 64; ++d) hv += hidden[d] * vsh[d];
      ws[WS_C] = c;
      ws[WS_SHID] = (hv + c) * INV_S;
    }
  }
}

// One block = 2 batch elements. 256 threads = 8 wave32.
__global__ __launch_bounds__(256)
void attn_kernel(const float* __restrict__ emb,    // (8192,8,64,9)
                 const float* __restrict__ own_b,  // 256
                 const float* __restrict__ ref_b,  // 256
                 const float* __restrict__ hidden, // 64
                 const float* __restrict__ ws,
                 float* __restrict__ out)          // (8192,16,64)
{
  __shared__ float sE[2][16][66];    // total_emb rows: 0-7 own, 8-15 enemy_ref
  __shared__ float sQK[2][16][260];  // [0]=own GEMM out, [1]=ref GEMM out (16x256)
  __shared__ float sW[2][16][17];    // softmax weights (16x16)

  const int tid  = threadIdx.x;
  const int lane = tid & 31;
  const int wv   = tid >> 5;     // wave id 0..7
  const int idx  = lane & 15;    // fragment row/col index
  const int hi   = lane >> 4;    // fragment half
  const long b0  = (long)blockIdx.x * 2;

  // ---------- Stage 1: own rows -> sE rows 0..7 (both batches) ----------
  #pragma unroll
  for (int r = 0; r < 4; ++r) {
    int f  = r * 256 + tid;            // 0..1023
    int b2 = f >> 9;
    int i  = (f >> 6) & 7;
    int d  = f & 63;
    sE[b2][i][d] = emb[(((b0 + b2) * 8 + i) * 64 + d) * 9];  // [...,0]
  }

  // ---------- Stage 2: enemy_ref rows -> sE rows 8..15 ----------
  const float c0   = ws[WS_C];
  const float shid = ws[WS_SHID];
  const int   d0   = 2 * lane;                       // this lane owns dims d0,d0+1
  const float vx = ws[WS_V + d0], vy = ws[WS_V + d0 + 1];
  const float hx = hidden[d0],    hy = hidden[d0 + 1];

  #pragma unroll
  for (int pp = 0; pp < 2; ++pp) {
    int p  = wv + pp * 8;              // pair 0..15
    int b2 = p >> 3;
    int e  = p & 7;
    const float* bb = emb + ((b0 + b2) * 8) * 576;
    int fl  = e * 64 + d0;             // flat index in sliced (D,8) block
    int dor = fl >> 3;
    int eor = fl & 7;
    float rx[8], ry[8], part[8];
    #pragma unroll
    for (int a = 0; a < 8; ++a) {      // agent rows of this (b,e)
      const float* pr = bb + (a * 576 + dor * 9 + 1 + eor);
      rx[a] = pr[0];
      ry[a] = pr[1];
      part[a] = rx[a] * vx + ry[a] * vy;
    }
    #pragma unroll
    for (int a = 0; a < 8; ++a) {      // full-wave butterfly sums
      float s = part[a];
      s += __shfl_xor(s, 1);  s += __shfl_xor(s, 2);  s += __shfl_xor(s, 4);
      s += __shfl_xor(s, 8);  s += __shfl_xor(s, 16);
      part[a] = s;
    }
    float sc[9];
    sc[0] = shid;
    #pragma unroll
    for (int a = 0; a < 8; ++a) sc[a + 1] = (part[a] + c0) * INV_S;
    float m = sc[0];
    #pragma unroll
    for (int i2 = 1; i2 < 9; ++i2) m = fmaxf(m, sc[i2]);
    float pw[9], Z = 0.f;
    #pragma unroll
    for (int i2 = 0; i2 < 9; ++i2) { pw[i2] = __expf(sc[i2] - m); Z += pw[i2]; }
    float rZ = 1.0f / Z;
    float o0 = pw[0] * hx, o1 = pw[0] * hy;
    #pragma unroll
    for (int a = 0; a < 8; ++a) { o0 += pw[a + 1] * rx[a]; o1 += pw[a + 1] * ry[a]; }
    sE[b2][8 + e][d0]     = o0 * rZ;
    sE[b2][8 + e][d0 + 1] = o1 * rZ;
  }
  __syncthreads();

  // ---------- Stage 3: QK GEMMs (16x64 @ 64x256), own + ref ----------
  // tile T: g = weight set (0 own rows / 1 ref rows), n0 = 16-col tile
  #pragma unroll
  for (int tt = 0; tt < 4; ++tt) {
    int T  = wv * 4 + tt;              // 0..31
    int g  = T >> 4;
    int n0 = (T & 15) * 16;
    const float* arow = &sE[idx >> 3][(g ? 8 : 0) + (idx & 7)][0];
    const float* brow = ws + (g ? WS_WTREF : WS_WTOWN) + (n0 + idx) * 64;
    v8f acc0 = {}, acc1 = {};
    #pragma unroll
    for (int kk = 0; kk < 64; kk += 8) {
      int k2 = kk + 2 * hi;
      v2f a0 = { arow[k2], arow[k2 + 1] };
      v2f b0f = *(const v2f*)(brow + k2);
      acc0 = wmma_f32(a0, b0f, acc0);
      v2f a1 = { arow[k2 + 4], arow[k2 + 5] };
      v2f b1f = *(const v2f*)(brow + k2 + 4);
      acc1 = wmma_f32(a1, b1f, acc1);
    }
    const float* bptr = g ? ref_b : own_b;
    float bias = bptr[n0 + idx];
    #pragma unroll
    for (int r = 0; r < 8; ++r)
      sQK[g][r + 8 * hi][n0 + idx] = acc0[r] + acc1[r] + bias;
  }
  __syncthreads();

  // ---------- Stage 4: energy (16x16, K=128) + softmax over rows ----------
  if (wv < 2) {
    int b2 = wv;
    const float* qrow = &sQK[idx >> 3][b2 * 8 + (idx & 7)][0];    // q cols 0..127
    const float* krow = &sQK[idx >> 3][b2 * 8 + (idx & 7)][128];  // k cols 0..127
    v8f acc0 = {}, acc1 = {};
    #pragma unroll
    for (int kk = 0; kk < 128; kk += 8) {
      int k2 = kk + 2 * hi;
      v2f a0 = *(const v2f*)(qrow + k2);
      v2f b0f = *(const v2f*)(krow + k2);
      acc0 = wmma_f32(a0, b0f, acc0);
      v2f a1 = *(const v2f*)(qrow + k2 + 4);
      v2f b1f = *(const v2f*)(krow + k2 + 4);
      acc1 = wmma_f32(a1, b1f, acc1);
    }
    float e[8];
    #pragma unroll
    for (int r = 0; r < 8; ++r) e[r] = (acc0[r] + acc1[r]) * INV_S;
    float m8 = e[0];
    #pragma unroll
    for (int r = 1; r < 8; ++r) m8 = fmaxf(m8, e[r]);
    float M = fmaxf(m8, __shfl_xor(m8, 16));     // combine row-halves (col j=idx)
    float p[8], s8 = 0.f;
    #pragma unroll
    for (int r = 0; r < 8; ++r) { p[r] = __expf(e[r] - M); s8 += p[r]; }
    float Z = s8 + __shfl_xor(s8, 16);
    float rZ = 1.0f / Z;
    #pragma unroll
    for (int r = 0; r < 8; ++r) sW[b2][r + 8 * hi][idx] = p[r] * rZ;
  }
  __syncthreads();

  // ---------- Stage 5: out = w^T @ total_emb  (16x16 @ 16x64) ----------
  {
    int b2 = wv >> 2;
    int n0 = (wv & 3) * 16;
    v8f acc = {};
    #pragma unroll
    for (int kk = 0; kk < 16; kk += 4) {
      int k2 = kk + 2 * hi;
      v2f a = { sW[b2][k2][idx],      sW[b2][k2 + 1][idx] };
      v2f b = { sE[b2][k2][n0 + idx], sE[b2][k2 + 1][n0 + idx] };
      acc = wmma_f32(a, b, acc);
    }
    float* op = out + (b0 + b2) * 1024 + n0 + idx;
    #pragma unroll
    for (int r = 0; r < 8; ++r)
      op[(r + 8 * hi) * 64] = acc[r];
  }
}

extern "C" void kernel_launch(void* const* d_in, const int* in_sizes, int n_in,
                              void* d_out, int out_size, void* d_ws, size_t ws_size,
                              hipStream_t stream) {
  const float* emb    = (const float*)d_in[0];
  const float* own_w  = (const float*)d_in[1];
  const float* own_b  = (const float*)d_in[2];
  const float* ene_w  = (const float*)d_in[3];
  const float* ene_b  = (const float*)d_in[4];
  const float* ref_w  = (const float*)d_in[5];
  const float* ref_b  = (const float*)d_in[6];
  const float* hidden = (const float*)d_in[7];
  float* ws  = (float*)d_ws;   // needs >= 131336 bytes
  float* out = (float*)d_out;

  prep_kernel<<<64, 256, 0, stream>>>(own_w, ref_w, ene_w, ene_b, hidden, ws);
  attn_kernel<<<4096, 256, 0, stream>>>(emb, own_b, ref_b, hidden, ws, out);
}